// GCN_71433896067547
// MI455X (gfx1250) — compile-verified
//
#include <hip/hip_runtime.h>
#include <hip/hip_bf16.h>

typedef __attribute__((ext_vector_type(2))) float v2f;
typedef __attribute__((ext_vector_type(8))) float v8f;

#define NN     50000
#define EE     800000
#define FIN    128
#define HIDD   128
#define NCLS   10
#define BN_EPS 1e-5f

// ---------------------------------------------------------------- degree
__global__ __launch_bounds__(256) void deg_kernel(const int* __restrict__ ei,
                                                  float* __restrict__ deg, int nE) {
    int e = blockIdx.x * 256 + threadIdx.x;
    if (e < nE) atomicAdd(&deg[ei[nE + e]], 1.0f);   // dst row of edge_index
}

__global__ __launch_bounds__(256) void dinv_kernel(const float* __restrict__ deg,
                                                   float* __restrict__ dinv, int n) {
    int i = blockIdx.x * 256 + threadIdx.x;
    if (i < n) {
        float d = deg[i] + 1.0f;                     // +1 self loop
        dinv[i] = (d > 0.0f) ? rsqrtf(d) : 0.0f;
    }
}

// ---------------------------------------------------------------- GEMM1: h1 = x @ W1  (fp32 WMMA 16x16x4)
// LDS layout: K-pairs interleaved so each lane's two B values (rows kk,kk+1,
// same column) are contiguous -> single aligned ds_load_b64, no reg shuffles.
//   ldsW[(k>>1)*256 + col*2 + (k&1)] = W1[k][col]
__global__ __launch_bounds__(256) void gemm_x_w1(const float* __restrict__ x,
                                                 const float* __restrict__ W1,
                                                 float* __restrict__ h1, int n) {
    __shared__ float ldsW[FIN * HIDD];               // 64 KB
    for (int i = threadIdx.x; i < FIN * HIDD; i += 256) {
        int k = i >> 7, c = i & 127;
        ldsW[(k >> 1) * 256 + c * 2 + (k & 1)] = W1[i];
    }
    __syncthreads();

    const int wave = threadIdx.x >> 5;
    const int lane = threadIdx.x & 31;
    const int half = lane >> 4;
    const int l    = lane & 15;
    const int m0   = blockIdx.x * 128 + wave * 16;   // 8 waves x 16 rows
    const int row  = m0 + l;
    const int rowc = row < n ? row : n - 1;          // clamp loads; stores predicated
    const float* xrow = x + (size_t)rowc * FIN;

    v8f acc[8] = {};                                 // 8 N-tiles of 16 cols
    for (int k = 0; k < FIN; k += 4) {
        const int kk = k + 2 * half;                 // even
        v2f a = *(const v2f*)(xrow + kk);            // A: row m0+l, K = kk,kk+1
        const float* bbase = ldsW + (kk >> 1) * 256 + l * 2;
        #pragma unroll
        for (int t = 0; t < 8; ++t) {
            v2f b = *(const v2f*)(bbase + t * 32);   // B rows kk,kk+1; col t*16+l
            acc[t] = __builtin_amdgcn_wmma_f32_16x16x4_f32(
                false, a, false, b, (short)0, acc[t], false, false);
        }
    }
    #pragma unroll
    for (int t = 0; t < 8; ++t) {
        #pragma unroll
        for (int r = 0; r < 8; ++r) {
            int orow = m0 + r + 8 * half;            // D: VGPR r -> M=r / r+8
            if (orow < n) h1[(size_t)orow * HIDD + t * 16 + l] = acc[t][r];
        }
    }
}

// ---------------------------------------------------------------- edge scatter, 128 features: one wave per edge
__global__ __launch_bounds__(256) void aggregate128(const float* __restrict__ h,
                                                    const int* __restrict__ ei,
                                                    const float* __restrict__ dinv,
                                                    float* __restrict__ out,
                                                    int nE, int n) {
    int gw   = (blockIdx.x * 256 + threadIdx.x) >> 5;  // global wave = edge id
    int lane = threadIdx.x & 31;
    int ET   = nE + n;
    if (gw >= ET) return;
    int s, d;
    if (gw < nE) { s = ei[gw]; d = ei[nE + gw]; }
    else         { s = d = gw - nE; }                  // self loops
    float nm = dinv[s] * dinv[d];
    const float4 v = *(const float4*)(h + (size_t)s * HIDD + lane * 4);
    float* o = out + (size_t)d * HIDD + lane * 4;
    atomicAdd(o + 0, v.x * nm);
    atomicAdd(o + 1, v.y * nm);
    atomicAdd(o + 2, v.z * nm);
    atomicAdd(o + 3, v.w * nm);
}

// ---------------------------------------------------------------- BN statistics
__global__ __launch_bounds__(128) void bn_stats(const float* __restrict__ agg,
                                                const float* __restrict__ b1,
                                                float* __restrict__ stats, int n) {
    int t = threadIdx.x;                              // feature 0..127
    float bs = b1[t];
    float s = 0.0f, q = 0.0f;
    for (int r = blockIdx.x; r < n; r += gridDim.x) {
        float v = agg[(size_t)r * HIDD + t] + bs;
        s += v; q += v * v;
    }
    atomicAdd(&stats[t], s);
    atomicAdd(&stats[HIDD + t], q);
}

__global__ __launch_bounds__(128) void bn_finalize(float* __restrict__ stats, int n) {
    int t = threadIdx.x;
    float invn = 1.0f / (float)n;
    float mu   = stats[t] * invn;
    float var  = stats[HIDD + t] * invn - mu * mu;
    stats[t]        = mu;
    stats[HIDD + t] = rsqrtf(var + BN_EPS);
}

__global__ __launch_bounds__(256) void bn_apply_relu(const float* __restrict__ agg,
                                                     const float* __restrict__ b1,
                                                     const float* __restrict__ gamma,
                                                     const float* __restrict__ beta,
                                                     const float* __restrict__ stats,
                                                     float* __restrict__ hbn, int n) {
    size_t idx = (size_t)blockIdx.x * 256 + threadIdx.x;
    if (idx >= (size_t)n * HIDD) return;
    int t = (int)(idx & (HIDD - 1));
    float v = (agg[idx] + b1[t] - stats[t]) * stats[HIDD + t] * gamma[t] + beta[t];
    hbn[idx] = fmaxf(v, 0.0f);
}

// ---------------------------------------------------------------- GEMM2: h2 = hbn @ W2 (N padded 10->16)
// Same K-pair LDS layout: ldsW[(k>>1)*32 + col*2 + (k&1)]
__global__ __launch_bounds__(256) void gemm_h_w2(const float* __restrict__ hbn,
                                                 const float* __restrict__ W2,
                                                 float* __restrict__ h2, int n) {
    __shared__ float ldsW[HIDD * 16];                 // 8 KB, zero-padded cols
    for (int i = threadIdx.x; i < HIDD * 16; i += 256) {
        int k = i >> 4, c = i & 15;
        ldsW[(k >> 1) * 32 + c * 2 + (k & 1)] = (c < NCLS) ? W2[k * NCLS + c] : 0.0f;
    }
    __syncthreads();

    const int wave = threadIdx.x >> 5;
    const int lane = threadIdx.x & 31;
    const int half = lane >> 4;
    const int l    = lane & 15;
    const int m0   = blockIdx.x * 128 + wave * 16;
    const int row  = m0 + l;
    const int rowc = row < n ? row : n - 1;
    const float* hrow = hbn + (size_t)rowc * HIDD;

    v8f acc = {};
    for (int k = 0; k < HIDD; k += 4) {
        const int kk = k + 2 * half;                  // even
        v2f a = *(const v2f*)(hrow + kk);
        v2f b = *(const v2f*)(ldsW + (kk >> 1) * 32 + l * 2);
        acc = __builtin_amdgcn_wmma_f32_16x16x4_f32(
            false, a, false, b, (short)0, acc, false, false);
    }
    #pragma unroll
    for (int r = 0; r < 8; ++r) {
        int orow = m0 + r + 8 * half;
        if (orow < n && l < NCLS) h2[(size_t)orow * NCLS + l] = acc[r];
    }
}

// ---------------------------------------------------------------- edge scatter, 10 features: 16 threads per edge
__global__ __launch_bounds__(256) void aggregate10(const float* __restrict__ h2,
                                                   const int* __restrict__ ei,
                                                   const float* __restrict__ dinv,
                                                   float* __restrict__ out,
                                                   int nE, int n) {
    long long idx = (long long)blockIdx.x * 256 + threadIdx.x;
    int f = (int)(idx & 15);
    long long e = idx >> 4;
    int ET = nE + n;
    if (e >= ET || f >= NCLS) return;
    int s, d;
    if (e < nE) { s = ei[e]; d = ei[nE + e]; }
    else        { s = d = (int)(e - nE); }
    float nm = dinv[s] * dinv[d];
    atomicAdd(out + (size_t)d * NCLS + f, h2[(size_t)s * NCLS + f] * nm);
}

// ---------------------------------------------------------------- + b2, log_softmax
__global__ __launch_bounds__(256) void logsoftmax_kernel(const float* __restrict__ agg2,
                                                         const float* __restrict__ b2,
                                                         float* __restrict__ out, int n) {
    int i = blockIdx.x * 256 + threadIdx.x;
    if (i >= n) return;
    float v[NCLS];
    float m = -3.0e38f;
    #pragma unroll
    for (int j = 0; j < NCLS; ++j) {
        v[j] = agg2[(size_t)i * NCLS + j] + b2[j];
        m = fmaxf(m, v[j]);
    }
    float s = 0.0f;
    #pragma unroll
    for (int j = 0; j < NCLS; ++j) s += expf(v[j] - m);
    float lse = m + logf(s);
    #pragma unroll
    for (int j = 0; j < NCLS; ++j) out[(size_t)i * NCLS + j] = v[j] - lse;
}

// ---------------------------------------------------------------- host launch
extern "C" void kernel_launch(void* const* d_in, const int* in_sizes, int n_in,
                              void* d_out, int out_size, void* d_ws, size_t ws_size,
                              hipStream_t stream) {
    const float* x     = (const float*)d_in[0];
    const int*   ei    = (const int*)d_in[1];      // [2][E], row0=src, row1=dst
    const float* W1    = (const float*)d_in[2];
    const float* b1    = (const float*)d_in[3];
    const float* gamma = (const float*)d_in[4];
    const float* beta  = (const float*)d_in[5];
    const float* W2    = (const float*)d_in[6];
    const float* b2    = (const float*)d_in[7];
    float*       out   = (float*)d_out;

    const int n  = NN;
    const int nE = EE;

    // workspace carve-out (~56 MB)
    float* ws   = (float*)d_ws;
    float* deg  = ws;                       ws += n;                  // [N]
    float* dinv = ws;                       ws += n;                  // [N]
    float* h1   = ws;                       ws += (size_t)n * HIDD;   // [N,128]
    float* agg1 = ws;                       ws += (size_t)n * HIDD;   // [N,128]
    float* h2   = ws;                       ws += (size_t)n * NCLS;   // [N,10]
    float* agg2 = ws;                       ws += (size_t)n * NCLS;   // [N,10]
    float* stats= ws;                       ws += 2 * HIDD;           // [256]

    hipMemsetAsync(deg,   0, sizeof(float) * n,                 stream);
    hipMemsetAsync(agg1,  0, sizeof(float) * (size_t)n * HIDD,  stream);
    hipMemsetAsync(agg2,  0, sizeof(float) * (size_t)n * NCLS,  stream);
    hipMemsetAsync(stats, 0, sizeof(float) * 2 * HIDD,          stream);

    // 1. degrees + dinv
    deg_kernel <<<(nE + 255) / 256, 256, 0, stream>>>(ei, deg, nE);
    dinv_kernel<<<(n  + 255) / 256, 256, 0, stream>>>(deg, dinv, n);

    // 2. h1 = x @ W1  (fp32 WMMA)
    gemm_x_w1<<<(n + 127) / 128, 256, 0, stream>>>(x, W1, h1, n);

    // 3. agg1 = scatter_add(h1[src] * norm)
    {
        long long waves = (long long)nE + n;               // one wave per edge
        int blocks = (int)((waves * 32 + 255) / 256);
        aggregate128<<<blocks, 256, 0, stream>>>(h1, ei, dinv, agg1, nE, n);
    }

    // 4. batch-norm + relu
    bn_stats    <<<512, 128, 0, stream>>>(agg1, b1, stats, n);
    bn_finalize <<<1,   128, 0, stream>>>(stats, n);
    bn_apply_relu<<<(int)(((size_t)n * HIDD + 255) / 256), 256, 0, stream>>>(
        agg1, b1, gamma, beta, stats, h1 /* reuse as hbn */, n);

    // 5. h2 = hbn @ W2  (fp32 WMMA, padded N)
    gemm_h_w2<<<(n + 127) / 128, 256, 0, stream>>>(h1, W2, h2, n);

    // 6. agg2 = scatter_add(h2[src] * norm)
    {
        long long work = ((long long)nE + n) * 16;
        int blocks = (int)((work + 255) / 256);
        aggregate10<<<blocks, 256, 0, stream>>>(h2, ei, dinv, agg2, nE, n);
    }

    // 7. + b2 and log-softmax
    logsoftmax_kernel<<<(n + 255) / 256, 256, 0, stream>>>(agg2, b2, out, n);
}